// RGCN_80762565034486
// MI455X (gfx1250) — compile-verified
//
#include <hip/hip_runtime.h>
#include <hip/hip_bf16.h>

// ---------------------------------------------------------------------------
// RGCN (2 layers, basis decomposition, mean aggregation) for MI455X / gfx1250.
//
//   per layer:
//     Wbig[k][col] = concat over r of W[r]  (+ root as strip 8), built once.
//     strip 8 (root):  out = x @ root + bias            (WMMA GEMM, full write)
//     for r = 0..7:    Hr  = x @ W[r]                   (WMMA GEMM, 25.6MB, L2)
//                      out[dst] += Hr[src] * inv_deg    (bucketed edge scatter,
//                                                        atomics hit L2)
//   ReLU (between layers) fused into layer-1 GEMM A-loads as a template
//   parameter (no runtime branch in the hot loop). Degrees/buckets once.
//
// GEMM: the 128x128 Wbig strip (64KB) is staged into LDS once per block via
// async global->LDS (ASYNCcnt + s_wait_asynccnt), B fragments come from the
// DS pipe (ds_load_2addr), each wave owns two 16-row M-tiles to double B
// reuse per v_wmma_f32_16x16x4_f32.
// ---------------------------------------------------------------------------

typedef float v2f __attribute__((ext_vector_type(2)));
typedef float v8f __attribute__((ext_vector_type(8)));
typedef int   v4i __attribute__((ext_vector_type(4)));

typedef __attribute__((address_space(1))) v4i* gas_v4i_p;   // global (AS1)
typedef __attribute__((address_space(3))) v4i* las_v4i_p;   // LDS (AS3)

#define DD    128          // feature dim (in = hid = out)
#define RNUM  8            // relations
#define BNUM  8            // bases
#define LDW   1152         // (RNUM+1)*DD : columns of Wbig (strips 0..7 = W[r], 8 = root)
#define LDS_S 136          // padded LDS row stride (dwords); 2*136 % 64 == 16 -> no bank
                           // overlap between the two half-wave K groups

// ---------------------------------------------------------------------------
__global__ void rgcn_build_w(const float* __restrict__ basis,
                             const float* __restrict__ comp,
                             const float* __restrict__ root,
                             float* __restrict__ Wbig) {
    int col = blockIdx.y * blockDim.x + threadIdx.x;
    int k   = blockIdx.x;
    if (col >= LDW) return;
    float v;
    if (col < RNUM * DD) {
        int r = col >> 7, o = col & 127;
        float acc = 0.f;
#pragma unroll
        for (int b = 0; b < BNUM; ++b)
            acc = fmaf(comp[r * BNUM + b], basis[(b * DD + k) * DD + o], acc);
        v = acc;
    } else {
        v = root[k * DD + (col - RNUM * DD)];
    }
    Wbig[k * LDW + col] = v;
}

// ---------------------------------------------------------------------------
__global__ void rgcn_deg(const int* __restrict__ ei, const int* __restrict__ et,
                         float* __restrict__ deg, int E) {
    int e = blockIdx.x * blockDim.x + threadIdx.x;
    if (e >= E) return;
    int dst = ei[E + e];
    int r   = et[e];
    unsafeAtomicAdd(&deg[dst * RNUM + r], 1.0f);
}

// --------------------- edge bucketing by relation (counting sort) ----------
__global__ void rgcn_count(const int* __restrict__ et, int* __restrict__ cnt, int E) {
    int e = blockIdx.x * blockDim.x + threadIdx.x;
    if (e < E) atomicAdd(&cnt[et[e]], 1);
}

__global__ void rgcn_scan(const int* __restrict__ cnt, int* __restrict__ roff,
                          int* __restrict__ cursor) {
    if (blockIdx.x == 0 && threadIdx.x == 0) {
        int acc = 0;
#pragma unroll
        for (int r = 0; r < RNUM; ++r) { roff[r] = acc; cursor[r] = acc; acc += cnt[r]; }
        roff[RNUM] = acc;
    }
}

__global__ void rgcn_bucket(const int* __restrict__ et, int* __restrict__ cursor,
                            int* __restrict__ sorted, int E) {
    int e = blockIdx.x * blockDim.x + threadIdx.x;
    if (e >= E) return;
    int p = atomicAdd(&cursor[et[e]], 1);
    sorted[p] = e;
}

// ---------------------------------------------------------------------------
// WMMA GEMM: dst[N,128] = act(X)[N,128] @ Wbig[:, strip*128 .. +128) (+bias)
// Block = 256 thr = 8 waves; wave owns TWO 16-row M-tiles x 128 cols.
template <int RELU_IN>
__global__ void __launch_bounds__(256)
rgcn_gemm_wmma(const float* __restrict__ X, const float* __restrict__ Wbig,
               float* __restrict__ dst, int strip, const float* __restrict__ bias,
               int Mtiles) {
    __shared__ float bsm[DD * LDS_S];            // 128 rows * 136 dwords = 68KB

    const int tid  = threadIdx.x;
    const int colb = strip * DD;

    // ---- stage 128x128 B strip into LDS (rows padded to LDS_S) ------------
#pragma unroll
    for (int i = 0; i < 16; ++i) {
        const int j   = tid + (i << 8);          // 4096 b128 transfers total
        const int row = j >> 5;                  // 0..127
        const int ch  = (j & 31) << 2;           // 0,4,...,124
        const float* gsrc = Wbig + (size_t)row * LDW + colb + ch;
        const int    loff = row * LDS_S + ch;
#if defined(__has_builtin) && __has_builtin(__builtin_amdgcn_global_load_async_to_lds_b128)
        __builtin_amdgcn_global_load_async_to_lds_b128(
            (gas_v4i_p)gsrc, (las_v4i_p)&bsm[loff], 0, 0);
#else
        *(float4*)&bsm[loff] = *(const float4*)gsrc;
#endif
    }
#if defined(__has_builtin) && __has_builtin(__builtin_amdgcn_global_load_async_to_lds_b128)
#if defined(__has_builtin) && __has_builtin(__builtin_amdgcn_s_wait_asynccnt)
    __builtin_amdgcn_s_wait_asynccnt(0);
#else
    asm volatile("s_wait_asynccnt 0" ::: "memory");
#endif
#endif
    __syncthreads();

    const int lane = tid & 31;
    const int wid  = tid >> 5;
    const int m0   = blockIdx.x * 16 + wid * 2;  // first of two M-tiles
    if (m0 >= Mtiles) return;                    // wave-uniform, after barrier
    const int m1   = m0 + 1;
    const int m1c  = (m1 < Mtiles) ? m1 : (Mtiles - 1);  // clamp loads

    const int coll  = lane & 15;                 // B/D column within 16-wide tile
    const int khalf = (lane >> 4) << 1;          // 0 or 2 : K sub-offset

    v8f c0[8], c1[8];
#pragma unroll
    for (int t = 0; t < 8; ++t) {
        c0[t] = (v8f){0.f,0.f,0.f,0.f,0.f,0.f,0.f,0.f};
        c1[t] = (v8f){0.f,0.f,0.f,0.f,0.f,0.f,0.f,0.f};
    }

    const float* xr0 = X + (size_t)(m0 * 16 + coll) * DD;
    const float* xr1 = X + (size_t)(m1c * 16 + coll) * DD;

    for (int k = 0; k < DD; k += 4) {
        // A fragments (16x4 f32): lane holds row=lane&15, cols k+khalf, +1
        v2f a0 = *(const v2f*)(xr0 + k + khalf);
        v2f a1 = *(const v2f*)(xr1 + k + khalf);
        if (RELU_IN) {                           // compile-time: no branch in loop
            a0.x = fmaxf(a0.x, 0.f); a0.y = fmaxf(a0.y, 0.f);
            a1.x = fmaxf(a1.x, 0.f); a1.y = fmaxf(a1.y, 0.f);
        }
        const int kr = k + khalf;
#pragma unroll
        for (int t = 0; t < 8; ++t) {
            // B fragment (4x16 f32) from LDS: lane holds col, rows kr, kr+1
            v2f b;
            b.x = bsm[kr * LDS_S + t * 16 + coll];
            b.y = bsm[(kr + 1) * LDS_S + t * 16 + coll];
            c0[t] = __builtin_amdgcn_wmma_f32_16x16x4_f32(
                false, a0, false, b, (short)0, c0[t], false, false);
            c1[t] = __builtin_amdgcn_wmma_f32_16x16x4_f32(
                false, a1, false, b, (short)0, c1[t], false, false);
        }
    }

    // D tile 16x16 f32: VGPR v -> row v + (lane>=16 ? 8 : 0), col = lane&15
    const int rsub = (lane >> 4) << 3;
#pragma unroll
    for (int t = 0; t < 8; ++t) {
        const float badd = bias ? bias[t * 16 + coll] : 0.f;
        float* d0 = dst + (size_t)(m0 * 16 + rsub) * DD + t * 16 + coll;
#pragma unroll
        for (int v = 0; v < 8; ++v)
            d0[(size_t)v * DD] = c0[t][v] + badd;
        if (m1 < Mtiles) {
            float* d1 = dst + (size_t)(m1 * 16 + rsub) * DD + t * 16 + coll;
#pragma unroll
            for (int v = 0; v < 8; ++v)
                d1[(size_t)v * DD] = c1[t][v] + badd;
        }
    }
}

// ---------------------------------------------------------------------------
// Scatter one relation's bucketed edges: out[dst] += Hr[src] * inv_deg.
__global__ void __launch_bounds__(256)
rgcn_scatter(const int* __restrict__ sorted, const int* __restrict__ roff, int r,
             const int* __restrict__ ei, int E,
             const float* __restrict__ deg, const float* __restrict__ Hr,
             float* __restrict__ out) {
    const int start = roff[r], end = roff[r + 1];
    const int i = start + blockIdx.x * 8 + (threadIdx.x >> 5);
    if (i >= end) return;
    const int e   = sorted[i];
    const int src = ei[e];
    const int dst = ei[E + e];
    const float inv = 1.0f / fmaxf(deg[dst * RNUM + r], 1.0f);
    const int lane = threadIdx.x & 31;
    const float4 v = *(const float4*)(Hr + (size_t)src * DD + lane * 4);
    float* o = out + (size_t)dst * DD + lane * 4;
    unsafeAtomicAdd(o + 0, v.x * inv);
    unsafeAtomicAdd(o + 1, v.y * inv);
    unsafeAtomicAdd(o + 2, v.z * inv);
    unsafeAtomicAdd(o + 3, v.w * inv);
}

// ---------------------------------------------------------------------------
static inline size_t align_up(size_t x) { return (x + 255) & ~(size_t)255; }

extern "C" void kernel_launch(void* const* d_in, const int* in_sizes, int n_in,
                              void* d_out, int out_size, void* d_ws, size_t ws_size,
                              hipStream_t stream) {
    (void)n_in; (void)out_size; (void)ws_size;

    const float* x  = (const float*)d_in[0];
    const int*   ei = (const int*)d_in[1];     // [2,E] src row then dst row
    const int*   et = (const int*)d_in[2];     // [E]
    const float* basisL[2] = {(const float*)d_in[3], (const float*)d_in[7]};
    const float* compL [2] = {(const float*)d_in[4], (const float*)d_in[8]};
    const float* rootL [2] = {(const float*)d_in[5], (const float*)d_in[9]};
    const float* biasL [2] = {(const float*)d_in[6], (const float*)d_in[10]};

    const int N = in_sizes[0] / DD;
    const int E = in_sizes[2];

    // workspace carve-out
    char* p = (char*)d_ws;
    float* Wbig = (float*)p;  p += align_up((size_t)DD * LDW * sizeof(float));
    float* Hr   = (float*)p;  p += align_up((size_t)N * DD * sizeof(float));
    float* h0   = (float*)p;  p += align_up((size_t)N * DD * sizeof(float));
    float* deg  = (float*)p;  p += align_up((size_t)N * RNUM * sizeof(float));
    int* cnt    = (int*)p;    p += align_up(RNUM * sizeof(int));
    int* roff   = (int*)p;    p += align_up((RNUM + 1) * sizeof(int));
    int* cursor = (int*)p;    p += align_up(RNUM * sizeof(int));
    int* sorted = (int*)p;    p += align_up((size_t)E * sizeof(int));

    const int Mtiles    = (N + 15) / 16;
    const dim3 gemmGrid((Mtiles + 15) / 16);      // 16 M-tiles per block (8 waves x 2)
    const int  eThreads = 256;
    const int  eBlocks  = (E + eThreads - 1) / eThreads;
    const int  sBlocks  = (E + 7) / 8;            // worst case: all edges one relation

    // ---- graph-constant preprocessing (deg + relation buckets), once -------
    (void)hipMemsetAsync(deg, 0, (size_t)N * RNUM * sizeof(float), stream);
    (void)hipMemsetAsync(cnt, 0, RNUM * sizeof(int), stream);
    rgcn_deg   <<<eBlocks, eThreads, 0, stream>>>(ei, et, deg, E);
    rgcn_count <<<eBlocks, eThreads, 0, stream>>>(et, cnt, E);
    rgcn_scan  <<<1, 32, 0, stream>>>(cnt, roff, cursor);
    rgcn_bucket<<<eBlocks, eThreads, 0, stream>>>(et, cursor, sorted, E);

    // ---- two RGCN layers ---------------------------------------------------
    for (int l = 0; l < 2; ++l) {
        const float* xin  = (l == 0) ? x  : h0;
        float*       outl = (l == 0) ? h0 : (float*)d_out;

        rgcn_build_w<<<dim3(DD, (LDW + 255) / 256), 256, 0, stream>>>(
            basisL[l], compL[l], rootL[l], Wbig);

        if (l == 0) {
            // self-loop + bias initializes the accumulator (full write)
            rgcn_gemm_wmma<0><<<gemmGrid, 256, 0, stream>>>(
                x, Wbig, outl, RNUM, biasL[l], Mtiles);
            for (int r = 0; r < RNUM; ++r) {
                rgcn_gemm_wmma<0><<<gemmGrid, 256, 0, stream>>>(
                    x, Wbig, Hr, r, nullptr, Mtiles);
                rgcn_scatter<<<sBlocks, 256, 0, stream>>>(
                    sorted, roff, r, ei, E, deg, Hr, outl);
            }
        } else {
            // ReLU of h0 fused into the A-operand loads (compile-time)
            rgcn_gemm_wmma<1><<<gemmGrid, 256, 0, stream>>>(
                xin, Wbig, outl, RNUM, biasL[l], Mtiles);
            for (int r = 0; r < RNUM; ++r) {
                rgcn_gemm_wmma<1><<<gemmGrid, 256, 0, stream>>>(
                    xin, Wbig, Hr, r, nullptr, Mtiles);
                rgcn_scatter<<<sBlocks, 256, 0, stream>>>(
                    sorted, roff, r, ei, E, deg, Hr, outl);
            }
        }
    }
}